// SPDBatchNormImpl_53626961657878
// MI455X (gfx1250) — compile-verified
//
#include <hip/hip_runtime.h>
#include <math.h>

// ---------------- problem constants ----------------
#define B_SZ 512
#define C_SZ 8
#define N_SZ 64
#define MAT (N_SZ * N_SZ)           // 4096 floats per matrix
#define CH_STRIDE (C_SZ * MAT)      // 32768 floats per "all channels" slab
#define ETA 1.0f
#define EPS_C 1e-5f
#define EIG_EPS 1e-6f
#define NSWEEP 6
#define BLK 128                     // 4 wave32 per matrix

// ---------------- workspace (float) layout ----------------
#define SLOT(k, c) ((size_t)(k) * CH_STRIDE + (size_t)(c) * MAT)
#define S_BM 0
#define S_BMSQ 1
#define S_BMISQ 2
#define S_GT 3
#define S_RM 4
#define S_RMISQ 5
#define S_GT2 6
#define S_MEANSQ 7
#define S_TMPA 8
#define S_TMPB 9
#define OFF_SSQ (10 * CH_STRIDE)
#define OFF_S (OFF_SSQ + C_SZ)

// ---------------- LDS (float index) layout ----------------
#define SM_A 0
#define SM_V 4096
#define SM_T1 8192
#define SM_T2 12288
#define SM_CS 16384
#define SM_SN 16416
#define SM_W 16448
#define SM_RED 16512   // 128 floats
#define SM_INT 16640   // 64 ints: pp[32], qq[32]
#define SMEM_BYTES ((16640 + 64) * 4)

typedef float v2f __attribute__((ext_vector_type(2)));
typedef float v8f __attribute__((ext_vector_type(8)));

struct Smem {
  float *A, *V, *T1, *T2;
  float *cs, *sn, *w, *red;
  int *pp, *qq;
};

__device__ inline Smem get_smem() {
  extern __shared__ float sm[];
  Smem s;
  s.A = sm + SM_A;
  s.V = sm + SM_V;
  s.T1 = sm + SM_T1;
  s.T2 = sm + SM_T2;
  s.cs = sm + SM_CS;
  s.sn = sm + SM_SN;
  s.w = sm + SM_W;
  s.red = sm + SM_RED;
  s.pp = (int*)(sm + SM_INT);
  s.qq = s.pp + 32;
  return s;
}

// ---------------- CDNA5 async global<->LDS tile movers ----------------
// 64x64 f32 tile = 16 KB = 1024 x b128 transfers; 8 per thread at BLK=128.
// LDS byte address = low 32 bits of the flat pointer (ISA aperture mapping).
__device__ inline void async_load_tile(float* __restrict__ dstLds,
                                       const float* __restrict__ src, int tid) {
  const unsigned base = (unsigned)(unsigned long long)(uintptr_t)dstLds;
  for (int i = tid; i < 1024; i += BLK) {
    unsigned loff = base + (unsigned)i * 16u;
    unsigned long long g = (unsigned long long)(uintptr_t)(src + i * 4);
    asm volatile("global_load_async_to_lds_b128 %0, %1, off"
                 :: "v"(loff), "v"(g) : "memory");
  }
  asm volatile("s_wait_asynccnt 0" ::: "memory");
  __syncthreads();
}

__device__ inline void async_store_tile(float* __restrict__ dstGlobal,
                                        const float* __restrict__ srcLds, int tid) {
  const unsigned base = (unsigned)(unsigned long long)(uintptr_t)srcLds;
  for (int i = tid; i < 1024; i += BLK) {
    unsigned loff = base + (unsigned)i * 16u;
    unsigned long long g = (unsigned long long)(uintptr_t)(dstGlobal + i * 4);
    asm volatile("global_store_async_from_lds_b128 %1, %0, off"
                 :: "v"(loff), "v"(g) : "memory");
  }
  asm volatile("s_wait_asynccnt 0" ::: "memory");
}

// ---------------- 64x64 f32 matmul on the matrix pipes ----------------
// D = Am * Bm (or Am * Bm^T). 128 threads = 4 wave32; each wave owns 4 of the
// sixteen 16x16 output tiles, 16 K-steps of V_WMMA_F32_16X16X4_F32 per tile.
__device__ inline void mm64(float* __restrict__ Dm, const float* __restrict__ Am,
                            const float* __restrict__ Bm, int tid, bool transB) {
  __syncthreads();
  const int wave = tid >> 5;
  const int lane = tid & 31;
  const int lane15 = lane & 15;
  const int half = lane >> 4;  // K-half / M-half selector per ISA layout
  for (int t = wave * 4; t < wave * 4 + 4; ++t) {
    const int ti = t >> 2;
    const int tj = t & 3;
    const int m = ti * 16 + lane15;
    const int n = tj * 16 + lane15;
    v8f acc = {};
    for (int k4 = 0; k4 < 16; ++k4) {
      const int kb = k4 * 4 + half * 2;
      v2f a, b;
      a.x = Am[m * 64 + kb];
      a.y = Am[m * 64 + kb + 1];
      if (transB) {
        b.x = Bm[n * 64 + kb];
        b.y = Bm[n * 64 + kb + 1];
      } else {
        b.x = Bm[kb * 64 + n];
        b.y = Bm[(kb + 1) * 64 + n];
      }
      acc = __builtin_amdgcn_wmma_f32_16x16x4_f32(false, a, false, b,
                                                  (short)0, acc, false, false);
    }
    // C/D layout: lanes 0-15 hold rows ti*16+0..7 at col n, lanes 16-31 rows +8
    for (int r = 0; r < 8; ++r)
      Dm[(ti * 16 + half * 8 + r) * 64 + n] = acc[r];
  }
  __syncthreads();
}

// ---------------- parallel cyclic Jacobi eigensolver ----------------
// A (LDS) -> near-diagonal, V -> eigenvectors (columns). 128 threads: the two
// half-blocks each apply 16 of the 32 disjoint Givens rotations per step.
__device__ inline void jacobi64(Smem s, int tid) {
  for (int j = tid; j < MAT; j += BLK) s.V[j] = 0.f;
  __syncthreads();
  if (tid < 64) s.V[tid * 65] = 1.f;
  __syncthreads();
  const int k0 = (tid >> 6) * 16;  // rotation subrange for this half-block
  for (int sweep = 0; sweep < NSWEEP; ++sweep) {
    for (int r = 0; r < 63; ++r) {
      if (tid < 32) {
        // round-robin tournament pairing: position 0 fixed, others rotate
        int i0 = tid, i1 = 63 - tid;
        int p = (i0 == 0) ? 0 : ((i0 - 1 + r) % 63) + 1;
        int q = ((i1 - 1 + r) % 63) + 1;
        if (p > q) { int tt = p; p = q; q = tt; }
        float apq = s.A[p * 64 + q];
        float c = 1.f, sv = 0.f;
        if (fabsf(apq) > 1e-12f) {
          float theta = (s.A[q * 64 + q] - s.A[p * 64 + p]) / (2.f * apq);
          float tt = copysignf(1.f, theta) / (fabsf(theta) + sqrtf(theta * theta + 1.f));
          c = rsqrtf(tt * tt + 1.f);
          sv = tt * c;
        }
        s.cs[tid] = c;
        s.sn[tid] = sv;
        s.pp[tid] = p;
        s.qq[tid] = q;
      }
      __syncthreads();
      {  // row phase: thread owns column j; rotations touch disjoint rows
        const int j = tid & 63;
        for (int k = k0; k < k0 + 16; ++k) {
          const int p = s.pp[k], q = s.qq[k];
          const float c = s.cs[k], sv = s.sn[k];
          float ap = s.A[p * 64 + j], aq = s.A[q * 64 + j];
          s.A[p * 64 + j] = c * ap + sv * aq;
          s.A[q * 64 + j] = -sv * ap + c * aq;
        }
      }
      __syncthreads();
      {  // column phase + eigenvector update: thread owns row i
        const int i = tid & 63;
        for (int k = k0; k < k0 + 16; ++k) {
          const int p = s.pp[k], q = s.qq[k];
          const float c = s.cs[k], sv = s.sn[k];
          float ap = s.A[i * 64 + p], aq = s.A[i * 64 + q];
          s.A[i * 64 + p] = c * ap + sv * aq;
          s.A[i * 64 + q] = -sv * ap + c * aq;
          float vp = s.V[i * 64 + p], vq = s.V[i * 64 + q];
          s.V[i * 64 + p] = c * vp + sv * vq;
          s.V[i * 64 + q] = -sv * vp + c * vq;
        }
      }
      __syncthreads();
    }
  }
  if (tid < 64) s.w[tid] = s.A[tid * 65];  // eigenvalues on the diagonal
  __syncthreads();
}

// T1[i][j] = V[i][j] * f(w[j]); modes: 0 sqrt, 1 invsqrt, 2 log, 3 exp, 4 pow(.,sp)
__device__ inline void scale_cols(Smem s, int mode, float sp, int tid) {
  for (int j = tid; j < MAT; j += BLK) {
    const float x = s.w[j & 63];
    const float xc = fmaxf(x, EIG_EPS);
    float y;
    switch (mode) {
      case 0: y = sqrtf(xc); break;
      case 1: y = rsqrtf(xc); break;
      case 2: y = logf(xc); break;
      case 3: y = expf(x); break;
      default: y = powf(xc, sp); break;
    }
    s.T1[j] = s.V[j] * y;
  }
  __syncthreads();
}

__device__ inline void cpy64(float* d, const float* a, int tid) {
  for (int j = tid; j < MAT; j += BLK) d[j] = a[j];
  __syncthreads();
}

__device__ inline float block_sum(float v, float* red, int tid) {
  red[tid] = v;
  __syncthreads();
  for (int st = BLK / 2; st > 0; st >>= 1) {
    if (tid < st) red[tid] += red[tid + st];
    __syncthreads();
  }
  float r = red[0];
  __syncthreads();
  return r;
}

// ---------------- kernel 0: batch mean over B + zero accumulators ----------------
__global__ void k_batch_mean(const float* __restrict__ X, float* __restrict__ ws) {
  const int idx = blockIdx.x * blockDim.x + threadIdx.x;  // over C*4096
  if (idx < CH_STRIDE) {
    float acc = 0.f;
    for (int b = 0; b < B_SZ; ++b) acc += X[(size_t)b * CH_STRIDE + idx];
    ws[SLOT(S_BM, 0) + idx] = acc * (1.f / (float)B_SZ);
    ws[SLOT(S_GT, 0) + idx] = 0.f;
  }
  if (idx < 2 * C_SZ) ws[OFF_SSQ + idx] = 0.f;
}

// ---------------- kernel 1: per-channel bm^{1/2}, bm^{-1/2} ----------------
__global__ void k_channel_sqrt(float* __restrict__ ws) {
  Smem s = get_smem();
  const int c = blockIdx.x;
  const int tid = threadIdx.x;
  cpy64(s.A, ws + SLOT(S_BM, c), tid);
  jacobi64(s, tid);
  scale_cols(s, 0, 0.f, tid);                 // sqrt
  mm64(s.T2, s.T1, s.V, tid, true);           // V f(w) V^T
  cpy64(ws + SLOT(S_BMSQ, c), s.T2, tid);
  scale_cols(s, 1, 0.f, tid);                 // invsqrt
  mm64(s.T2, s.T1, s.V, tid, true);
  cpy64(ws + SLOT(S_BMISQ, c), s.T2, tid);
}

// ---------------- kernel 2: pass 1 — log map at batch mean ----------------
__global__ void k_pass1(const float* __restrict__ X, float* __restrict__ ws) {
  Smem s = get_smem();
  const int b = blockIdx.x >> 3;
  const int c = blockIdx.x & 7;
  const int tid = threadIdx.x;
  async_load_tile(s.T2, X + ((size_t)b * C_SZ + c) * MAT, tid);  // X via ASYNCcnt
  cpy64(s.T1, ws + SLOT(S_BMISQ, c), tid);              // S
  mm64(s.A, s.T1, s.T2, tid, false);                    // S X
  mm64(s.T2, s.A, s.T1, tid, false);                    // S X S
  cpy64(s.A, s.T2, tid);
  jacobi64(s, tid);
  scale_cols(s, 2, 0.f, tid);                           // log
  mm64(s.T2, s.T1, s.V, tid, true);                     // XT
  float loc = 0.f;
  for (int j = tid; j < MAT; j += BLK) {
    const float v = s.T2[j];
    atomicAdd(&ws[SLOT(S_GT, c) + j], v);
    loc += v * v;
  }
  loc = block_sum(loc, s.red, tid);
  if (tid == 0) atomicAdd(&ws[OFF_SSQ + c], loc);
}

// ---------------- kernel 3: channel-level chain ----------------
__global__ void k_channel_stats(const float* __restrict__ mean,
                                const float* __restrict__ stdv,
                                const float* __restrict__ rmean,
                                const float* __restrict__ rvar,
                                float* __restrict__ ws) {
  Smem s = get_smem();
  const int c = blockIdx.x;
  const int tid = threadIdx.x;
  // --- tangent mean GT (normalize accumulated sum) ---
  for (int j = tid; j < MAT; j += BLK) {
    const float g = ws[SLOT(S_GT, c) + j] * (1.f / (float)B_SZ);
    ws[SLOT(S_GT, c) + j] = g;
    s.A[j] = g;
  }
  __syncthreads();
  // --- batch_mean = bm_sq expm(GT) bm_sq ---
  jacobi64(s, tid);
  scale_cols(s, 3, 0.f, tid);                 // exp
  mm64(s.T2, s.T1, s.V, tid, true);           // expm(GT)
  cpy64(s.T1, ws + SLOT(S_BMSQ, c), tid);
  mm64(s.A, s.T1, s.T2, tid, false);
  mm64(s.T2, s.A, s.T1, tid, false);
  cpy64(ws + SLOT(S_TMPA, c), s.T2, tid);     // batch_mean
  // --- geodesic(running_mean, batch_mean, ETA) ---
  cpy64(s.A, rmean + (size_t)c * MAT, tid);
  jacobi64(s, tid);
  scale_cols(s, 0, 0.f, tid);                 // Asq
  mm64(s.T2, s.T1, s.V, tid, true);
  cpy64(ws + SLOT(S_RM, c), s.T2, tid);       // Asq (temp in rm slot)
  scale_cols(s, 1, 0.f, tid);                 // Ainvsq
  mm64(s.T2, s.T1, s.V, tid, true);
  cpy64(ws + SLOT(S_TMPB, c), s.T2, tid);
  cpy64(s.T1, ws + SLOT(S_TMPB, c), tid);     // Ainvsq
  cpy64(s.T2, ws + SLOT(S_TMPA, c), tid);     // batch_mean
  mm64(s.A, s.T1, s.T2, tid, false);
  mm64(s.T2, s.A, s.T1, tid, false);          // Mid
  cpy64(s.A, s.T2, tid);
  jacobi64(s, tid);
  scale_cols(s, 4, ETA, tid);                 // Mid^ETA
  mm64(s.T2, s.T1, s.V, tid, true);
  cpy64(s.T1, ws + SLOT(S_RM, c), tid);       // Asq
  mm64(s.A, s.T1, s.T2, tid, false);
  mm64(s.T2, s.A, s.T1, tid, false);          // rm
  cpy64(ws + SLOT(S_RM, c), s.T2, tid);
  // --- rm^{-1/2} ---
  cpy64(s.A, s.T2, tid);
  jacobi64(s, tid);
  scale_cols(s, 1, 0.f, tid);
  mm64(s.T2, s.T1, s.V, tid, true);
  cpy64(ws + SLOT(S_RMISQ, c), s.T2, tid);
  // --- GT2 = logm(bm_invsq rm bm_invsq) ---
  cpy64(s.T1, ws + SLOT(S_BMISQ, c), tid);
  cpy64(s.T2, ws + SLOT(S_RM, c), tid);
  mm64(s.A, s.T1, s.T2, tid, false);
  mm64(s.T2, s.A, s.T1, tid, false);
  cpy64(s.A, s.T2, tid);
  jacobi64(s, tid);
  scale_cols(s, 2, 0.f, tid);
  mm64(s.T2, s.T1, s.V, tid, true);
  cpy64(ws + SLOT(S_GT2, c), s.T2, tid);
  // --- mean^{1/2} ---
  cpy64(s.A, mean + (size_t)c * MAT, tid);
  jacobi64(s, tid);
  scale_cols(s, 0, 0.f, tid);
  mm64(s.T2, s.T1, s.V, tid, true);
  cpy64(ws + SLOT(S_MEANSQ, c), s.T2, tid);
  // --- dispersion scalar: var = E||XT||^2 - 2<GT,GT2> + ||GT2||^2 ---
  float dg = 0.f, gg = 0.f;
  for (int j = tid; j < MAT; j += BLK) {
    const float g = ws[SLOT(S_GT, c) + j];
    const float g2 = ws[SLOT(S_GT2, c) + j];
    dg += g * g2;
    gg += g2 * g2;
  }
  dg = block_sum(dg, s.red, tid);
  gg = block_sum(gg, s.red, tid);
  if (tid == 0) {
    const float bv = ws[OFF_SSQ + c] * (1.f / (float)B_SZ) - 2.f * dg + gg;
    const float rv = (1.f - ETA) * rvar[c] + ETA * bv;
    ws[OFF_S + c] = stdv[c] * rsqrtf(rv + EPS_C);
  }
}

// ---------------- kernel 4: pass 2 — rescale dispersion, transport to `mean` ----------------
__global__ void k_pass2(const float* __restrict__ X, const float* __restrict__ ws,
                        float* __restrict__ out) {
  Smem s = get_smem();
  const int b = blockIdx.x >> 3;
  const int c = blockIdx.x & 7;
  const int tid = threadIdx.x;
  const float sp = ws[OFF_S + c];
  async_load_tile(s.T2, X + ((size_t)b * C_SZ + c) * MAT, tid);  // X via ASYNCcnt
  cpy64(s.T1, ws + SLOT(S_RMISQ, c), tid);
  mm64(s.A, s.T1, s.T2, tid, false);
  mm64(s.T2, s.A, s.T1, tid, false);          // rm^-1/2 X rm^-1/2
  cpy64(s.A, s.T2, tid);
  jacobi64(s, tid);
  scale_cols(s, 4, sp, tid);                  // w^s
  mm64(s.T2, s.T1, s.V, tid, true);           // powm
  cpy64(s.T1, ws + SLOT(S_MEANSQ, c), tid);
  mm64(s.A, s.T1, s.T2, tid, false);
  mm64(s.T2, s.A, s.T1, tid, false);          // mean_sq . mean_sq
  async_store_tile(out + ((size_t)b * C_SZ + c) * MAT, s.T2, tid);
}

extern "C" void kernel_launch(void* const* d_in, const int* in_sizes, int n_in,
                              void* d_out, int out_size, void* d_ws, size_t ws_size,
                              hipStream_t stream) {
  (void)in_sizes; (void)n_in; (void)out_size; (void)ws_size;
  const float* X = (const float*)d_in[0];
  const float* mean = (const float*)d_in[1];
  const float* stdv = (const float*)d_in[2];
  const float* rmean = (const float*)d_in[3];
  const float* rvar = (const float*)d_in[4];
  float* out = (float*)d_out;
  float* ws = (float*)d_ws;

  k_batch_mean<<<(CH_STRIDE + 255) / 256, 256, 0, stream>>>(X, ws);
  k_channel_sqrt<<<C_SZ, BLK, SMEM_BYTES, stream>>>(ws);
  k_pass1<<<B_SZ * C_SZ, BLK, SMEM_BYTES, stream>>>(X, ws);
  k_channel_stats<<<C_SZ, BLK, SMEM_BYTES, stream>>>(mean, stdv, rmean, rvar, ws);
  k_pass2<<<B_SZ * C_SZ, BLK, SMEM_BYTES, stream>>>(X, ws, out);
}